// MultiMaskAttentionLayer_7095285973097
// MI455X (gfx1250) — compile-verified
//
#include <hip/hip_runtime.h>
#include <cmath>

typedef __attribute__((ext_vector_type(2))) float v2f;
typedef __attribute__((ext_vector_type(8))) float v8f;

constexpr long  L     = 262144;  // 64^3 voxels
constexpr int   NB    = 2;       // batch
constexpr int   DIM   = 64;
constexpr float EPS_F = 1e-6f;
constexpr int   NSPAN = 64;          // spans per batch for deterministic attention partials
constexpr long  SPANV = L / NSPAN;   // 4096 voxels per span

__device__ __forceinline__ float gelu_exact(float x) {
    return 0.5f * x * (1.0f + erff(x * 0.70710678118654752440f));
}

// ---------------------------------------------------------------------------
// Pointwise 16x16 channel-mix GEMM per 16-voxel tile via V_WMMA_F32_16X16X4_F32,
// + bias + channel LayerNorm (+ optional exact GELU).
//   out[b][cbaseOut+o][vox] = act( LN_o( sum_c W[o][c]*x[b][c][vox] + bias[o] ) )
// One wave = one 16-voxel tile; block = 4 waves = 64 voxels.
// ---------------------------------------------------------------------------
__global__ void pw16_ln(const float* __restrict__ x,
                        const float* __restrict__ W,      // [16][16] row-major (o,c)
                        const float* __restrict__ bias,
                        const float* __restrict__ gamma,
                        const float* __restrict__ beta,
                        float* __restrict__ out,
                        long inBatchStride, long outBatchStride,
                        int cbaseOut, int doGelu)
{
    const int lane = threadIdx.x & 31;
    const int wave = threadIdx.x >> 5;   // 0..3
    const int n    = lane & 15;          // voxel within tile / D-matrix column
    const int hi   = lane >> 4;          // 0 or 1 (lane half)

    long gid = (long)blockIdx.x * 64 + wave * 16;   // global voxel id over B*L
    long b   = gid / L;
    long vox = gid - b * L;                         // 16-aligned tile base
    const float* xb = x + b * inBatchStride;

    v8f acc = {};
#pragma unroll
    for (int kk = 0; kk < 4; ++kk) {
        const int k0 = kk * 4 + 2 * hi;             // K index pair for this lane half
        v2f a, bv;
        a.x  = W[n * 16 + k0];                      // A[M=n][K=k0]
        a.y  = W[n * 16 + k0 + 1];
        bv.x = xb[(long)k0 * L + vox + n];          // B[K=k0][N=n]
        bv.y = xb[(long)(k0 + 1) * L + vox + n];
        acc  = __builtin_amdgcn_wmma_f32_16x16x4_f32(false, a, false, bv,
                                                     (short)0, acc, false, false);
    }

    // lane (n,hi) holds channels {r+8*hi : r=0..7} of voxel vox+n
    float vals[8];
    float s = 0.f, ss = 0.f;
#pragma unroll
    for (int r = 0; r < 8; ++r) {
        const int ch = r + 8 * hi;
        float v = acc[r] + bias[ch];
        vals[r] = v; s += v; ss += v * v;
    }
    s  += __shfl_xor(s, 16, 32);    // combine the two channel halves of this voxel
    ss += __shfl_xor(ss, 16, 32);
    const float mean = s  * (1.0f / 16.0f);
    const float var  = ss * (1.0f / 16.0f) - mean * mean;
    const float inv  = rsqrtf(var + EPS_F);

    float* ob = out + b * outBatchStride;
#pragma unroll
    for (int r = 0; r < 8; ++r) {
        const int ch = r + 8 * hi;
        float v = (vals[r] - mean) * inv * gamma[ch] + beta[ch];
        if (doGelu) v = gelu_exact(v);
        ob[(long)(cbaseOut + ch) * L + vox + n] = v;
    }
}

// ---------------------------------------------------------------------------
// Pointwise 4x4 channel-mix + bias + LN (+ optional GELU, + optional identity add).
// One thread per voxel. Used for the 4-channel query_map / out_project blocks.
// ---------------------------------------------------------------------------
__global__ void pw4_ln(const float* __restrict__ x, const float* __restrict__ W,
                       const float* __restrict__ bias, const float* __restrict__ g,
                       const float* __restrict__ be, const float* __restrict__ identity,
                       float* __restrict__ out, int doGelu)
{
    long gid = (long)blockIdx.x * blockDim.x + threadIdx.x;   // over B*L exactly
    long b = gid / L, vox = gid - b * L;
    const float* xb = x + b * 4 * L;
    float xi[4];
#pragma unroll
    for (int c = 0; c < 4; ++c) xi[c] = xb[c * L + vox];
    float y[4]; float s = 0.f, ss = 0.f;
#pragma unroll
    for (int o = 0; o < 4; ++o) {
        float v = bias[o];
#pragma unroll
        for (int c = 0; c < 4; ++c) v = fmaf(W[o * 4 + c], xi[c], v);
        y[o] = v; s += v; ss += v * v;
    }
    const float mean = s * 0.25f;
    const float var  = ss * 0.25f - mean * mean;
    const float inv  = rsqrtf(var + EPS_F);
    float* ob = out + b * 4 * L;
#pragma unroll
    for (int o = 0; o < 4; ++o) {
        float v = (y[o] - mean) * inv * g[o] + be[o];
        if (doGelu) v = gelu_exact(v);
        if (identity) v += identity[b * 4 * L + o * L + vox];
        ob[o * L + vox] = v;
    }
}

// ---------------------------------------------------------------------------
// Depthwise 3x3x3 conv (SAME, zero pad) + bias + LN + GELU. Thread per voxel.
// ---------------------------------------------------------------------------
template <int C>
__global__ void dwconv_ln_gelu(const float* __restrict__ x,
                               const float* __restrict__ w2,  // [C][27]
                               const float* __restrict__ b2,
                               const float* __restrict__ g,
                               const float* __restrict__ be,
                               float* __restrict__ out)
{
    long gid = (long)blockIdx.x * blockDim.x + threadIdx.x;
    long b = gid / L, vox = gid - b * L;
    const int h = (int)(vox >> 12);
    const int w = (int)((vox >> 6) & 63);
    const int d = (int)(vox & 63);
    const float* xb = x + b * (long)C * L;

    float acc[C];
#pragma unroll
    for (int c = 0; c < C; ++c) acc[c] = b2[c];

    for (int i = 0; i < 3; ++i) {
        const int hh = h + i - 1;
        if (hh < 0 || hh >= DIM) continue;
        for (int j = 0; j < 3; ++j) {
            const int ww = w + j - 1;
            if (ww < 0 || ww >= DIM) continue;
            for (int kq = 0; kq < 3; ++kq) {
                const int dd = d + kq - 1;
                if (dd < 0 || dd >= DIM) continue;
                const long nv = ((long)hh << 12) | (ww << 6) | dd;
                const int widx = i * 9 + j * 3 + kq;
#pragma unroll
                for (int c = 0; c < C; ++c)
                    acc[c] = fmaf(w2[c * 27 + widx], xb[(long)c * L + nv], acc[c]);
            }
        }
    }
    float s = 0.f, ss = 0.f;
#pragma unroll
    for (int c = 0; c < C; ++c) { s += acc[c]; ss += acc[c] * acc[c]; }
    const float mean = s / (float)C;
    const float var  = ss / (float)C - mean * mean;
    const float inv  = rsqrtf(var + EPS_F);
    float* ob = out + b * (long)C * L;
#pragma unroll
    for (int c = 0; c < C; ++c) {
        float v = (acc[c] - mean) * inv * g[c] + be[c];
        ob[(long)c * L + vox] = gelu_exact(v);
    }
}

// ---------------------------------------------------------------------------
// Attention scores: partial[b][span][q][k] = sum_{vox in span} q[b][q][vox]*k[b][k][vox]
// via V_WMMA_F32_16X16X4_F32 with K = 4 voxels per issue.
// A = q tile (M = q-channel, rows 4..15 zero), B = k tile (N = 16 k-channels per wave).
// Block = 4 waves (k-channel groups 0..3); grid = B * NSPAN. Deterministic (no atomics).
// ---------------------------------------------------------------------------
__global__ void attn_score(const float* __restrict__ q, const float* __restrict__ k,
                           float* __restrict__ partial)
{
    const int lane = threadIdx.x & 31;
    const int wave = threadIdx.x >> 5;   // k-channel group
    const int n    = lane & 15;
    const int hi   = lane >> 4;
    const int span = blockIdx.x % NSPAN;
    const int b    = blockIdx.x / NSPAN;
    const long v0  = (long)span * SPANV;

    const float* qrow = q + (long)b * 4 * L + (long)(n < 4 ? n : 0) * L;
    const float* krow = k + (long)b * 64 * L + (long)(wave * 16 + n) * L;

    v8f acc = {};
    for (long vv = 0; vv < SPANV; vv += 4) {
        const long off = v0 + vv + 2 * hi;
        v2f a, bv;
        const float qx = qrow[off], qy = qrow[off + 1];
        a.x  = (n < 4) ? qx : 0.0f;          // pad q rows 4..15 with zeros
        a.y  = (n < 4) ? qy : 0.0f;
        bv.x = krow[off];
        bv.y = krow[off + 1];
        acc  = __builtin_amdgcn_wmma_f32_16x16x4_f32(false, a, false, bv,
                                                     (short)0, acc, false, false);
    }
    if (hi == 0) {                            // D rows 0..3 = valid q channels
        const float scale = 1.0f / 512.0f;    // L^-0.5, L = 512^2
        float* pp = partial + (long)(b * NSPAN + span) * 256;
#pragma unroll
        for (int r = 0; r < 4; ++r)
            pp[r * 64 + wave * 16 + n] = acc[r] * scale;
    }
}

__global__ void attn_reduce(const float* __restrict__ partial, float* __restrict__ attn)
{
    const int t = threadIdx.x;        // 0..511 -> b*256 + (q*64+k)
    const int b = t >> 8, idx = t & 255;
    float s = 0.f;
    for (int sp = 0; sp < NSPAN; ++sp)
        s += partial[(long)(b * NSPAN + sp) * 256 + idx];
    attn[t] = s;
}

// Masked softmax over the 64 key channels. Grid = B*4 rows, 64 threads.
__global__ void attn_softmax(float* __restrict__ attn, const unsigned char* __restrict__ mask)
{
    const int row = blockIdx.x;       // b*4 + qi
    const int b   = row >> 2;
    const int t   = threadIdx.x;      // key channel
    __shared__ float red[64];

    const bool m = mask[b * 4 + (t >> 4)] != 0;   // kc//16 = 16 key channels per modality bit
    const float lg = attn[row * 64 + t];
    float v = m ? lg : -1e30f;
    red[t] = v; __syncthreads();
    for (int s = 32; s > 0; s >>= 1) { if (t < s) red[t] = fmaxf(red[t], red[t + s]); __syncthreads(); }
    const float mx = red[0]; __syncthreads();
    const float e = m ? expf(v - mx) : 0.0f;
    red[t] = e; __syncthreads();
    for (int s = 32; s > 0; s >>= 1) { if (t < s) red[t] += red[t + s]; __syncthreads(); }
    attn[row * 64 + t] = e / red[0];
}

// x[b][q][vox] = sum_k attn[b][q][k] * v[b][k][vox]. Thread per voxel; attn row in LDS.
__global__ void attn_apply(const float* __restrict__ attn, const float* __restrict__ v,
                           float* __restrict__ x)
{
    __shared__ float a_s[256];
    long gid = (long)blockIdx.x * 256 + threadIdx.x;
    long b = gid / L, vox = gid - b * L;
    a_s[threadIdx.x] = attn[b * 256 + threadIdx.x];
    __syncthreads();
    const float* vb = v + b * 64 * L;
    float a0 = 0.f, a1 = 0.f, a2 = 0.f, a3 = 0.f;
    for (int kc = 0; kc < 64; ++kc) {
        const float vv = vb[(long)kc * L + vox];
        a0 = fmaf(a_s[0 * 64 + kc], vv, a0);
        a1 = fmaf(a_s[1 * 64 + kc], vv, a1);
        a2 = fmaf(a_s[2 * 64 + kc], vv, a2);
        a3 = fmaf(a_s[3 * 64 + kc], vv, a3);
    }
    float* ob = x + b * 4 * L;
    ob[0 * L + vox] = a0; ob[1 * L + vox] = a1;
    ob[2 * L + vox] = a2; ob[3 * L + vox] = a3;
}

// ---------------------------------------------------------------------------
extern "C" void kernel_launch(void* const* d_in, const int* in_sizes, int n_in,
                              void* d_out, int out_size, void* d_ws, size_t ws_size,
                              hipStream_t stream)
{
    const float* query = (const float*)d_in[0];
    const float* modals[4] = {(const float*)d_in[1], (const float*)d_in[2],
                              (const float*)d_in[3], (const float*)d_in[4]};
    const unsigned char* mask = (const unsigned char*)d_in[5];
    // params: blocks [query_map, out_project, key_f, val_f, key_t1ce, val_t1ce,
    //                 key_t1, val_t1, key_t2, val_t2] x
    //         [w1,b1,g1,be1, w2,b2,g2,be2, w3,b3,g3,be3]
    auto P = [&](int blk, int idx) { return (const float*)d_in[6 + blk * 12 + idx]; };

    float* ws = (float*)d_ws;
    float* Kb = ws;  ws += (size_t)NB * 64 * L;
    float* Vb = ws;  ws += (size_t)NB * 64 * L;
    float* T1 = ws;  ws += (size_t)NB * 16 * L;
    float* T2 = ws;  ws += (size_t)NB * 16 * L;
    float* Qb = ws;  ws += (size_t)NB * 4 * L;
    float* Xb = ws;  ws += (size_t)NB * 4 * L;
    float* Pt = ws;  ws += (size_t)NB * NSPAN * 256;
    float* At = ws;  ws += (size_t)NB * 256;

    const int gridPW16 = (int)(NB * L / 64);    // 8192 blocks x 128 thr (4 waves)
    const int gridVox  = (int)(NB * L / 256);   // 2048 blocks x 256 thr
    const long sC16 = 16 * L, sC64 = 64 * L;

    // ---- query_map (C=4) -> Qb ----
    pw4_ln<<<gridVox, 256, 0, stream>>>(query, P(0,0), P(0,1), P(0,2), P(0,3), nullptr, T1, 1);
    dwconv_ln_gelu<4><<<gridVox, 256, 0, stream>>>(T1, P(0,4), P(0,5), P(0,6), P(0,7), T2);
    pw4_ln<<<gridVox, 256, 0, stream>>>(T2, P(0,8), P(0,9), P(0,10), P(0,11), nullptr, Qb, 0);

    // ---- key/value blocks (C=16, WMMA channel-mix) -> Kb, Vb ----
    for (int m = 0; m < 4; ++m) {
        const int kb = 2 + 2 * m, vb = 3 + 2 * m;
        pw16_ln<<<gridPW16, 128, 0, stream>>>(modals[m], P(kb,0), P(kb,1), P(kb,2), P(kb,3),
                                              T1, sC16, sC16, 0, 1);
        dwconv_ln_gelu<16><<<gridVox, 256, 0, stream>>>(T1, P(kb,4), P(kb,5), P(kb,6), P(kb,7), T2);
        pw16_ln<<<gridPW16, 128, 0, stream>>>(T2, P(kb,8), P(kb,9), P(kb,10), P(kb,11),
                                              Kb, sC16, sC64, m * 16, 0);

        pw16_ln<<<gridPW16, 128, 0, stream>>>(modals[m], P(vb,0), P(vb,1), P(vb,2), P(vb,3),
                                              T1, sC16, sC16, 0, 1);
        dwconv_ln_gelu<16><<<gridVox, 256, 0, stream>>>(T1, P(vb,4), P(vb,5), P(vb,6), P(vb,7), T2);
        pw16_ln<<<gridPW16, 128, 0, stream>>>(T2, P(vb,8), P(vb,9), P(vb,10), P(vb,11),
                                              Vb, sC16, sC64, m * 16, 0);
    }

    // ---- attention (WMMA scores, deterministic reduce, masked softmax, apply) ----
    attn_score<<<NB * NSPAN, 128, 0, stream>>>(Qb, Kb, Pt);
    attn_reduce<<<1, 512, 0, stream>>>(Pt, At);
    attn_softmax<<<NB * 4, 64, 0, stream>>>(At, mask);
    attn_apply<<<gridVox, 256, 0, stream>>>(At, Vb, Xb);

    // ---- out_project (C=4) + identity -> d_out ----
    pw4_ln<<<gridVox, 256, 0, stream>>>(Xb, P(1,0), P(1,1), P(1,2), P(1,3), nullptr, T1, 1);
    dwconv_ln_gelu<4><<<gridVox, 256, 0, stream>>>(T1, P(1,4), P(1,5), P(1,6), P(1,7), T2);
    pw4_ln<<<gridVox, 256, 0, stream>>>(T2, P(1,8), P(1,9), P(1,10), P(1,11), query,
                                        (float*)d_out, 0);

    (void)in_sizes; (void)n_in; (void)out_size; (void)ws_size;
}